// SampleAndAggregate_91079076479552
// MI455X (gfx1250) — compile-verified
//
#include <hip/hip_runtime.h>
#include <hip/hip_bf16.h>

// ---------------- problem constants ----------------
#define N_NODES   100000
#define FEAT      256     // FEAT_DIM
#define MAX_DEG   128
#define BATCH     512
#define S1F       10      // layer-2 fanout
#define S2F       25      // layer-1 fanout
#define HID       256
#define NODES2B   (2 * BATCH)          // 1024
#define NS1       (NODES2B * S1F)      // 10240
#define NOUT      (2 * HID)            // 512

typedef __attribute__((ext_vector_type(2))) float v2f;
typedef __attribute__((ext_vector_type(8))) float v8f;

// ---------------- index construction ----------------
__global__ __launch_bounds__(256) void build_idx_kernel(
    const int* __restrict__ adj, const int* __restrict__ b1,
    const int* __restrict__ b2, const int* __restrict__ col1,
    int* __restrict__ nodes, int* __restrict__ s1) {
  int i = blockIdx.x * blockDim.x + threadIdx.x;
  if (i < NODES2B) nodes[i] = (i < BATCH) ? b1[i] : b2[i - BATCH];
  if (i < NS1) {
    int n = i / S1F;
    int node = (n < BATCH) ? b1[n] : b2[n - BATCH];
    s1[i] = adj[(size_t)node * MAX_DEG + col1[i]];
  }
}

// ---------------- weight k-pair packing ----------------
// Wp[k/2][n][2] = { W[k][n], W[k+1][n] }  -> B fragment becomes one b64 load
__global__ __launch_bounds__(256) void pack_w_kernel(
    const float* __restrict__ W, float* __restrict__ Wp, int total) {
  int i = blockIdx.x * blockDim.x + threadIdx.x;
  if (i >= total) return;
  int k = i >> 8;        // / HID
  int n = i & (HID - 1);
  Wp[(size_t)(k >> 1) * (2 * HID) + n * 2 + (k & 1)] = W[i];
}

// ---------------- 2-hop gather + mean : n1mean[10240,256] ----------------
// 4 rows per block, float4 loads (b128), fully coalesced.
__global__ __launch_bounds__(256) void mean_h2_kernel(
    const float* __restrict__ feat, const int* __restrict__ adj,
    const int* __restrict__ s1, const int* __restrict__ col2,
    float* __restrict__ n1mean) {
  __shared__ int idx[4][S2F];
  const int rb = blockIdx.x * 4;
  const int t  = threadIdx.x;
  if (t < 4 * S2F) {
    int rr = rb + t / S2F;
    int j  = t % S2F;
    idx[t / S2F][j] = adj[(size_t)s1[rr] * MAX_DEG + col2[rr * S2F + j]];
  }
  __syncthreads();
  const int g = t >> 6;          // row within block (0..3)
  const int c = (t & 63) * 4;    // feature column
  float sx = 0.f, sy = 0.f, sz = 0.f, sw = 0.f;
#pragma unroll 5
  for (int j = 0; j < S2F; ++j) {
    const float4 f = *(const float4*)(feat + (size_t)idx[g][j] * FEAT + c);
    sx += f.x; sy += f.y; sz += f.z; sw += f.w;
  }
  float4 o;
  o.x = sx * (1.0f / S2F); o.y = sy * (1.0f / S2F);
  o.z = sz * (1.0f / S2F); o.w = sw * (1.0f / S2F);
  *(float4*)(n1mean + (size_t)(rb + g) * FEAT + c) = o;
}

// ---------------- 1-hop gather + mean : h0mean[1024,256] ----------------
__global__ __launch_bounds__(256) void mean_h1feat_kernel(
    const float* __restrict__ feat, const int* __restrict__ s1,
    float* __restrict__ h0mean) {
  __shared__ int idx[4][S1F];
  const int nb = blockIdx.x * 4;
  const int t  = threadIdx.x;
  if (t < 4 * S1F) {
    idx[t / S1F][t % S1F] = s1[(nb + t / S1F) * S1F + (t % S1F)];
  }
  __syncthreads();
  const int g = t >> 6;
  const int c = (t & 63) * 4;
  float sx = 0.f, sy = 0.f, sz = 0.f, sw = 0.f;
#pragma unroll
  for (int j = 0; j < S1F; ++j) {
    const float4 f = *(const float4*)(feat + (size_t)idx[g][j] * FEAT + c);
    sx += f.x; sy += f.y; sz += f.z; sw += f.w;
  }
  float4 o;
  o.x = sx * (1.0f / S1F); o.y = sy * (1.0f / S1F);
  o.z = sz * (1.0f / S1F); o.w = sw * (1.0f / S1F);
  *(float4*)(h0mean + (size_t)(nb + g) * FEAT + c) = o;
}

// ---------------- fused concat-GEMM with f32 WMMA ----------------
// C[M,512]: cols [0,256)  = act( gather(A0,idx0) @ W0 )
//           cols [256,512)= act( A1 @ W1 )
// Block: 256 threads = 8 waves; 32 rows x 512 cols per block.
// Wave w: branch = w/4, 64-col window = (w%4)*64, 2 row-tiles x 4 col-tiles.
#define KCHUNK  64
#define LDA     68   // padded LDS row stride -> conflict-free b64 reads
#define MTILE   32

__global__ __launch_bounds__(256) void sage_gemm_wmma_kernel(
    const float* __restrict__ A0, const int* __restrict__ idx0,
    const float* __restrict__ W0p,
    const float* __restrict__ A1, const float* __restrict__ W1p,
    float* __restrict__ C, int M, int K, int relu) {
  __shared__ float ldsA[2][MTILE * LDA];   // 17408 B

  const int tid  = threadIdx.x;
  const int wave = tid >> 5;
  const int lane = tid & 31;
  const int gm   = blockIdx.x * MTILE;

  // staging: thread t loads 8 consecutive floats of row (t/8), cols (t&7)*8
  const int srow = tid >> 3;        // 0..31
  const int scol = (tid & 7) * 8;   // 0..56
  const size_t r0 = (size_t)(idx0 ? idx0[gm + srow] : (gm + srow)) * (size_t)K;
  const size_t r1 = (size_t)(gm + srow) * (size_t)K;

  const int branch  = wave >> 2;                 // 0: self, 1: neighbor
  const float* Wp   = branch ? W1p : W0p;
  const int colbase = (wave & 3) * 64;

  const int kp = (lane >> 4) * 2;                // k-pair select (0 or 2)
  const int nl = lane & 15;

  v8f acc[2][4] = {};

  for (int k0 = 0; k0 < K; k0 += KCHUNK) {
    if (k0 + KCHUNK < K) {
      __builtin_prefetch(A0 + r0 + k0 + KCHUNK + scol, 0, 1);
      __builtin_prefetch(A1 + r1 + k0 + KCHUNK + scol, 0, 1);
    }
    __syncthreads();
    const float4 a0a = *(const float4*)(A0 + r0 + k0 + scol);
    const float4 a0b = *(const float4*)(A0 + r0 + k0 + scol + 4);
    const float4 a1a = *(const float4*)(A1 + r1 + k0 + scol);
    const float4 a1b = *(const float4*)(A1 + r1 + k0 + scol + 4);
    *(float4*)(&ldsA[0][srow * LDA + scol])     = a0a;
    *(float4*)(&ldsA[0][srow * LDA + scol + 4]) = a0b;
    *(float4*)(&ldsA[1][srow * LDA + scol])     = a1a;
    *(float4*)(&ldsA[1][srow * LDA + scol + 4]) = a1b;
    __syncthreads();

    const float* lA = ldsA[branch];
#pragma unroll
    for (int kk = 0; kk < KCHUNK; kk += 4) {
      const v2f af0 = *(const v2f*)(&lA[nl * LDA + kk + kp]);
      const v2f af1 = *(const v2f*)(&lA[(16 + nl) * LDA + kk + kp]);
      const float* wrow = Wp + (size_t)((k0 + kk + kp) >> 1) * (2 * HID) + nl * 2;
#pragma unroll
      for (int t = 0; t < 4; ++t) {
        const v2f bf = *(const v2f*)(wrow + (colbase + t * 16) * 2);
        acc[0][t] = __builtin_amdgcn_wmma_f32_16x16x4_f32(
            false, af0, false, bf, (short)0, acc[0][t], false, false);
        acc[1][t] = __builtin_amdgcn_wmma_f32_16x16x4_f32(
            false, af1, false, bf, (short)0, acc[1][t], false, false);
      }
    }
  }

  // write out: VGPR v -> row m = v + 8*(lane>>4), col n = lane&15
  const int mh = (lane >> 4) * 8;
  const int cn = branch * HID + colbase + nl;
#pragma unroll
  for (int rt = 0; rt < 2; ++rt) {
#pragma unroll
    for (int t = 0; t < 4; ++t) {
#pragma unroll
      for (int v = 0; v < 8; ++v) {
        float val = acc[rt][t][v];
        if (relu) val = fmaxf(val, 0.f);
        C[(size_t)(gm + rt * 16 + mh + v) * NOUT + cn + t * 16] = val;
      }
    }
  }
}

// ---------------- mean over S1 of h1new : h1nm[1024,512] ----------------
// 2 rows per block, float4 loads.
__global__ __launch_bounds__(256) void mean_h1new_kernel(
    const float* __restrict__ h1new, float* __restrict__ h1nm) {
  const int t = threadIdx.x;
  const int n = blockIdx.x * 2 + (t >> 7);
  const int c = (t & 127) * 4;
  float sx = 0.f, sy = 0.f, sz = 0.f, sw = 0.f;
#pragma unroll
  for (int j = 0; j < S1F; ++j) {
    const float4 f = *(const float4*)(h1new + (size_t)(n * S1F + j) * NOUT + c);
    sx += f.x; sy += f.y; sz += f.z; sw += f.w;
  }
  float4 o;
  o.x = sx * (1.0f / S1F); o.y = sy * (1.0f / S1F);
  o.z = sz * (1.0f / S1F); o.w = sw * (1.0f / S1F);
  *(float4*)(h1nm + (size_t)n * NOUT + c) = o;
}

// ---------------- row-wise L2 normalize into d_out ----------------
__global__ __launch_bounds__(128) void l2norm_kernel(
    const float* __restrict__ inp, float* __restrict__ out) {
  __shared__ float red[128];
  const int n = blockIdx.x;
  const int t = threadIdx.x;
  const float4 v = *(const float4*)(inp + (size_t)n * NOUT + t * 4);
  red[t] = v.x * v.x + v.y * v.y + v.z * v.z + v.w * v.w;
  __syncthreads();
  for (int s = 64; s > 0; s >>= 1) {
    if (t < s) red[t] += red[t + s];
    __syncthreads();
  }
  const float sc = rsqrtf(fmaxf(red[0], 1e-12f));
  float4 o;
  o.x = v.x * sc; o.y = v.y * sc; o.z = v.z * sc; o.w = v.w * sc;
  *(float4*)(out + (size_t)n * NOUT + t * 4) = o;
}

// ---------------- host-side launch ----------------
extern "C" void kernel_launch(void* const* d_in, const int* in_sizes, int n_in,
                              void* d_out, int out_size, void* d_ws, size_t ws_size,
                              hipStream_t stream) {
  (void)in_sizes; (void)n_in; (void)out_size; (void)ws_size;
  const float* features = (const float*)d_in[0];
  const float* Ws1      = (const float*)d_in[1];
  const float* Wn1      = (const float*)d_in[2];
  const float* Ws2      = (const float*)d_in[3];
  const float* Wn2      = (const float*)d_in[4];
  const int*   adj      = (const int*)d_in[5];
  const int*   batch1   = (const int*)d_in[6];
  const int*   batch2   = (const int*)d_in[7];
  const int*   col1     = (const int*)d_in[8];
  const int*   col2     = (const int*)d_in[9];
  float*       out      = (float*)d_out;

  // workspace carve-up (byte offsets, all 4 KB aligned)
  char* ws = (char*)d_ws;
  int*   s1      = (int*)(ws + 0);                  //  40 KB  [10240]
  int*   nodes   = (int*)(ws + 49152);              //   4 KB  [1024]
  float* n1mean  = (float*)(ws + 65536);            // 10 MB   [10240,256]
  float* h0mean  = (float*)(ws + 10551296);         //  1 MB   [1024,256]
  float* h1new   = (float*)(ws + 11599872);         // 20 MB   [10240,512]
  float* h0new   = (float*)(ws + 32571392);         //  2 MB   [1024,512]
  float* h1nm    = (float*)(ws + 34668544);         //  2 MB   [1024,512]
  float* outraw  = (float*)(ws + 36765696);         //  2 MB   [1024,512]
  float* Ws1p    = (float*)(ws + 38862848);         // 256 KB  packed weights
  float* Wn1p    = (float*)(ws + 39124992);
  float* Ws2p    = (float*)(ws + 39387136);         // 512 KB
  float* Wn2p    = (float*)(ws + 39911424);         // 512 KB (end ~40.4 MB)

  // indices + weight packing (independent)
  build_idx_kernel<<<(NS1 + 255) / 256, 256, 0, stream>>>(adj, batch1, batch2,
                                                          col1, nodes, s1);
  pack_w_kernel<<<(FEAT * HID) / 256, 256, 0, stream>>>(Ws1, Ws1p, FEAT * HID);
  pack_w_kernel<<<(FEAT * HID) / 256, 256, 0, stream>>>(Wn1, Wn1p, FEAT * HID);
  pack_w_kernel<<<(NOUT * HID) / 256, 256, 0, stream>>>(Ws2, Ws2p, NOUT * HID);
  pack_w_kernel<<<(NOUT * HID) / 256, 256, 0, stream>>>(Wn2, Wn2p, NOUT * HID);

  // gathered means
  mean_h2_kernel<<<NS1 / 4, 256, 0, stream>>>(features, adj, s1, col2, n1mean);
  mean_h1feat_kernel<<<NODES2B / 4, 256, 0, stream>>>(features, s1, h0mean);

  // layer 1 (relu, concat)
  sage_gemm_wmma_kernel<<<NODES2B / MTILE, 256, 0, stream>>>(
      features, nodes, Ws1p, h0mean, Wn1p, h0new, NODES2B, FEAT, 1);
  sage_gemm_wmma_kernel<<<NS1 / MTILE, 256, 0, stream>>>(
      features, s1, Ws1p, n1mean, Wn1p, h1new, NS1, FEAT, 1);

  // layer 2 (identity act, concat) + normalize
  mean_h1new_kernel<<<NODES2B / 2, 256, 0, stream>>>(h1new, h1nm);
  sage_gemm_wmma_kernel<<<NODES2B / MTILE, 256, 0, stream>>>(
      h0new, nullptr, Ws2p, h1nm, Wn2p, outraw, NODES2B, NOUT, 0);

  l2norm_kernel<<<NODES2B, 128, 0, stream>>>(outraw, out);
}